// FallLoss_31439160606939
// MI455X (gfx1250) — compile-verified
//
#include <hip/hip_runtime.h>
#include <math.h>

#define THRELD        0.5f
#define LOC_SCALE     1.4f
#define CONTAIN_SCALE 4.0f
#define L_NOOBJ       1.2f
#define CLASS_SCALE   0.5f

typedef __attribute__((ext_vector_type(2))) float v2f;
typedef __attribute__((ext_vector_type(8))) float v8f;

// Full-wave (32 lane) sum using the CDNA5 matrix engine.
// A (16x4 f32): lane L<16 supplies A[L][0]=v, A[L][1]=0; lane L>=16 supplies
// A[L-16][2]=v, A[L-16][3]=0. With B = ones, D[m][n] = v[m] + v[m+16].
// D layout: VGPR j holds r[j] in lanes 0-15 and r[j+8] in lanes 16-31, so
// summing the 8 D registers gives sum(r[0..7]) / sum(r[8..15]) per half-wave;
// one xor-16 shuffle completes the 32-lane sum (broadcast to all lanes).
// Requires EXEC == all ones at the call site (callers are fully convergent).
__device__ __forceinline__ float wave_sum32(float v) {
    v2f a; a[0] = v;    a[1] = 0.0f;
    v2f b; b[0] = 1.0f; b[1] = 1.0f;
    v8f c = {};
    v8f d = __builtin_amdgcn_wmma_f32_16x16x4_f32(
        /*neg_a=*/false, a, /*neg_b=*/false, b,
        /*c_mod=*/(short)0, c, /*reuse_a=*/false, /*reuse_b=*/false);
    float s = d[0] + d[1] + d[2] + d[3] + d[4] + d[5] + d[6] + d[7];
    s += __shfl_xor(s, 16, 32);
    return s;
}

// Per-cell loss components, accumulated per thread, reduced per block.
// comp order: 0=loc 1=contain 2=not_contain 3=nooobj 4=class
__global__ __launch_bounds__(256) void yolo_loss_partial(
    const float* __restrict__ pred, const float* __restrict__ targ,
    float* __restrict__ ws, int ncells) {
    float acc0 = 0.f, acc1 = 0.f, acc2 = 0.f, acc3 = 0.f, acc4 = 0.f;

    const int stride = gridDim.x * blockDim.x;
    for (int i = blockIdx.x * blockDim.x + threadIdx.x; i < ncells; i += stride) {
        const float4* p4 = (const float4*)(pred + (size_t)i * 12);
        const float4* t4 = (const float4*)(targ + (size_t)i * 12);
        float4 pA = p4[0], pB = p4[1], pC = p4[2];   // p[0..3], p[4..7], p[8..11]
        float4 tA = t4[0], tB = t4[1], tC = t4[2];

        const float coo = (tB.x > THRELD) ? 1.0f : 0.0f;  // t[4] > 0.5
        const float noo = 1.0f - coo;

        // ---- no-object confidence loss (channels 4, 9) ----
        float d0 = fmaxf(pB.x - THRELD + 0.3f, 0.0f) - tB.x;   // p[4], t[4]
        float d1 = fmaxf(pC.y - THRELD + 0.3f, 0.0f) - tC.y;   // p[9], t[9]
        acc3 += noo * (d0 * d0 + d1 * d1);

        // ---- IoU of both pred boxes vs target box 0 ----
        // target box 0: cx=tA.x cy=tA.y w=tA.z h=tA.w
        const float tx1 = tA.x - 0.5f * tA.z, ty1 = tA.y - 0.5f * tA.w;
        const float tx2 = tA.x + 0.5f * tA.z, ty2 = tA.y + 0.5f * tA.w;
        const float tarea = (tx2 - tx1) * (ty2 - ty1);

        // pred box 0: cx=pA.x cy=pA.y w=pA.z h=pA.w conf=pB.x
        float x1 = pA.x - 0.5f * pA.z, y1 = pA.y - 0.5f * pA.w;
        float x2 = pA.x + 0.5f * pA.z, y2 = pA.y + 0.5f * pA.w;
        float iw = fmaxf(fminf(x2, tx2) - fmaxf(x1, tx1), 0.0f);
        float ih = fmaxf(fminf(y2, ty2) - fmaxf(y1, ty1), 0.0f);
        float inter0 = iw * ih;
        float area0  = (x2 - x1) * (y2 - y1);
        float iou0   = inter0 / (area0 + tarea - inter0);

        // pred box 1: cx=pB.y cy=pB.z w=pB.w h=pC.x conf=pC.y
        x1 = pB.y - 0.5f * pB.w; y1 = pB.z - 0.5f * pC.x;
        x2 = pB.y + 0.5f * pB.w; y2 = pB.z + 0.5f * pC.x;
        iw = fmaxf(fminf(x2, tx2) - fmaxf(x1, tx1), 0.0f);
        ih = fmaxf(fminf(y2, ty2) - fmaxf(y1, ty1), 0.0f);
        float inter1 = iw * ih;
        float area1  = (x2 - x1) * (y2 - y1);
        float iou1   = inter1 / (area1 + tarea - inter1);

        const bool sel = iou1 > iou0;  // argmax (ties -> box 0)

        // responsible pred box / target box, and the not-responsible conf
        const float prx = sel ? pB.y : pA.x;
        const float pry = sel ? pB.z : pA.y;
        const float prw = sel ? pB.w : pA.z;
        const float prh = sel ? pC.x : pA.w;
        const float prc = sel ? pC.y : pB.x;
        const float pnc = sel ? pB.x : pC.y;

        const float trx = sel ? tB.y : tA.x;
        const float try_ = sel ? tB.z : tA.y;
        const float trw = sel ? tB.w : tA.z;
        const float trh = sel ? tC.x : tA.w;
        const float trc = sel ? tC.y : tB.x;

        // contain / loc / not-contain / class
        const float dc = prc - trc;
        acc1 += coo * dc * dc;

        const float dx = prx - trx, dy = pry - try_;
        const float dw = __builtin_sqrtf(prw) - __builtin_sqrtf(trw);
        const float dh = __builtin_sqrtf(prh) - __builtin_sqrtf(trh);
        acc0 += coo * (dx * dx + dy * dy + dw * dw + dh * dh);

        acc2 += coo * pnc * pnc;

        const float c0 = pC.z - tC.z, c1 = pC.w - tC.w;  // channels 10, 11
        acc4 += coo * (c0 * c0 + c1 * c1);
    }

    // wave reduction via WMMA (all threads convergent here)
    float s0 = wave_sum32(acc0);
    float s1 = wave_sum32(acc1);
    float s2 = wave_sum32(acc2);
    float s3 = wave_sum32(acc3);
    float s4 = wave_sum32(acc4);

    __shared__ float lds[8][5];
    const int wave = threadIdx.x >> 5;
    const int lane = threadIdx.x & 31;
    if (lane == 0) {
        lds[wave][0] = s0; lds[wave][1] = s1; lds[wave][2] = s2;
        lds[wave][3] = s3; lds[wave][4] = s4;
    }
    __syncthreads();
    if (threadIdx.x < 5) {
        float s = 0.0f;
        #pragma unroll
        for (int w = 0; w < 8; ++w) s += lds[w][threadIdx.x];
        ws[blockIdx.x * 5 + threadIdx.x] = s;
    }
}

__global__ __launch_bounds__(256) void yolo_loss_final(
    const float* __restrict__ ws, float* __restrict__ out, int nblocks) {
    float acc[5] = {0.f, 0.f, 0.f, 0.f, 0.f};
    for (int b = threadIdx.x; b < nblocks; b += 256) {
        #pragma unroll
        for (int c = 0; c < 5; ++c) acc[c] += ws[b * 5 + c];
    }

    float s[5];
    #pragma unroll
    for (int c = 0; c < 5; ++c) s[c] = wave_sum32(acc[c]);

    __shared__ float lds[8][5];
    __shared__ float parts[5];
    const int wave = threadIdx.x >> 5;
    const int lane = threadIdx.x & 31;
    if (lane == 0) {
        #pragma unroll
        for (int c = 0; c < 5; ++c) lds[wave][c] = s[c];
    }
    __syncthreads();
    if (threadIdx.x < 5) {
        float v = 0.0f;
        #pragma unroll
        for (int w = 0; w < 8; ++w) v += lds[w][threadIdx.x];
        const float scales[5] = {LOC_SCALE, CONTAIN_SCALE, 1.0f, L_NOOBJ, CLASS_SCALE};
        v *= scales[threadIdx.x];
        parts[threadIdx.x] = v;
        out[1 + threadIdx.x] = v;
    }
    __syncthreads();
    if (threadIdx.x == 0)
        out[0] = parts[0] + parts[1] + parts[2] + parts[3] + parts[4];
}

extern "C" void kernel_launch(void* const* d_in, const int* in_sizes, int n_in,
                              void* d_out, int out_size, void* d_ws, size_t ws_size,
                              hipStream_t stream) {
    const float* pred = (const float*)d_in[0];
    const float* targ = (const float*)d_in[1];
    float* out = (float*)d_out;
    float* ws  = (float*)d_ws;

    const int ncells = in_sizes[0] / 12;  // 64*128*128 = 1,048,576

    int nblocks = 2048;                   // 2 cells/thread at 256 threads
    const int maxblocks = (int)(ws_size / (5 * sizeof(float)));
    if (nblocks > maxblocks) nblocks = maxblocks > 0 ? maxblocks : 1;

    yolo_loss_partial<<<nblocks, 256, 0, stream>>>(pred, targ, ws, ncells);
    yolo_loss_final<<<1, 256, 0, stream>>>(ws, out, nblocks);
}